// DecoderGRU_73993696575773
// MI455X (gfx1250) — compile-verified
//
#include <hip/hip_runtime.h>
#include <math.h>
#include <stdint.h>

typedef __attribute__((ext_vector_type(2))) float v2f;
typedef __attribute__((ext_vector_type(8))) float v8f;

#define T_STEPS 256
#define E_DIM   512
#define H_DIM   1024
#define V_DIM   32000
#define REC_BLOCKS 128

#define BK      512      // K half staged per pass
#define BSTRIDE 516      // floats per LDS row (+4 pad: bank stride 4 mod 64 -> conflict-free)

// ---------------------------------------------------------------- init
__global__ __launch_bounds__(256) void init_kernel(const float* __restrict__ encoding,
                                                   float* __restrict__ Hbuf,
                                                   int* __restrict__ bar) {
  int tid = threadIdx.x;
  if (tid < 2) bar[tid] = 0;                    // grid-barrier count / generation
  for (int i = tid; i < H_DIM; i += 256) Hbuf[i] = encoding[i];   // h_0
}

// ---------------------------------------------------- x-side gate GEMM (WMMA f32)
// Xg[t,j] = emb[tok[t]] . Wgx[j] + bgx[j] + bgh[j]   for g in {z,r,h}
__global__ __launch_bounds__(256) void x_precompute_kernel(
    const int* __restrict__ tokens, const float* __restrict__ emb,
    const float* __restrict__ wzx, const float* __restrict__ bzx,
    const float* __restrict__ wrx, const float* __restrict__ brx,
    const float* __restrict__ whx, const float* __restrict__ bhx,
    const float* __restrict__ bzh, const float* __restrict__ brh,
    const float* __restrict__ bhh,
    float* __restrict__ Xz, float* __restrict__ Xr, float* __restrict__ Xh) {
  int gw   = (blockIdx.x * blockDim.x + threadIdx.x) >> 5;  // global wave id
  int gate = gw >> 10;                 // 1024 tiles (16 m-tiles x 64 n-tiles) per gate
  int rem  = gw & 1023;
  int mt   = rem >> 6;
  int nt   = rem & 63;
  int lane = threadIdx.x & 31;
  int mn   = lane & 15;
  int khalf = (lane >> 4) << 1;        // 0 or 2: upper half-wave handles K+2,K+3

  const float* W  = (gate == 0) ? wzx : (gate == 1) ? wrx : whx;
  const float* bx = (gate == 0) ? bzx : (gate == 1) ? brx : bhx;
  const float* bh = (gate == 0) ? bzh : (gate == 1) ? brh : bhh;
  float*       X  = (gate == 0) ? Xz  : (gate == 1) ? Xr  : Xh;

  int col = nt * 16 + mn;
  const float* Arow = emb + (long long)tokens[mt * 16 + mn] * E_DIM;
  const float* Brow = W   + (long long)col * E_DIM;          // B = W^T: contiguous in W row
  float cinit = bx[col] + bh[col];                            // bias depends on n only
  v8f c;
#pragma unroll
  for (int i = 0; i < 8; ++i) c[i] = cinit;

#pragma unroll 4
  for (int k = 0; k < E_DIM; k += 4) {
    v2f a = *(const v2f*)(Arow + k + khalf);
    v2f b = *(const v2f*)(Brow + k + khalf);
    c = __builtin_amdgcn_wmma_f32_16x16x4_f32(false, a, false, b, (short)0, c, false, false);
  }

  int mhi = (lane >> 4) * 8;           // C/D: upper half-wave holds rows M+8
#pragma unroll
  for (int i = 0; i < 8; ++i)
    X[(long long)(mt * 16 + i + mhi) * H_DIM + col] = c[i];
}

// ---------------------------------------------------------------- recurrent core
__device__ __forceinline__ float wave_sum(float v) {
#pragma unroll
  for (int off = 16; off > 0; off >>= 1) v += __shfl_xor(v, off, 32);
  return v;
}

__device__ __forceinline__ void grid_barrier(int* count, volatile int* gen, int nblk) {
  __threadfence();                      // publish this thread's writes device-wide
  __syncthreads();
  if (threadIdx.x == 0) {
    int g = *gen;
    if (atomicAdd(count, 1) == nblk - 1) {
      *count = 0;
      __threadfence();
      atomicAdd((int*)gen, 1);          // release everyone
    } else {
      while (*gen == g) __builtin_amdgcn_s_sleep(1);
    }
  }
  __syncthreads();
}

// 1024 wave32s; wave j owns output row j of all three recurrent GEMVs.
__global__ __launch_bounds__(256) void gru_recurrent_kernel(
    const float* __restrict__ Xz, const float* __restrict__ Xr, const float* __restrict__ Xh,
    const float* __restrict__ wzh, const float* __restrict__ wrh, const float* __restrict__ whh,
    float* Hbuf, float* zbuf, float* rhbuf, int* bar) {
  int j    = (blockIdx.x * blockDim.x + threadIdx.x) >> 5;   // 0..1023
  int lane = threadIdx.x & 31;
  const float* wz = wzh + (long long)j * H_DIM;
  const float* wr = wrh + (long long)j * H_DIM;
  const float* wh = whh + (long long)j * H_DIM;

  for (int t = 0; t < T_STEPS; ++t) {
    const float* h = Hbuf + (long long)t * H_DIM;
    // phase A: z and r rows (share the h loads)
    float az = 0.f, ar = 0.f;
    for (int k = lane; k < H_DIM; k += 32) {
      float hv = h[k];
      az += wz[k] * hv;
      ar += wr[k] * hv;
    }
    az = wave_sum(az);
    ar = wave_sum(ar);
    if (lane == 0) {
      float z = 1.f / (1.f + __expf(-(Xz[t * H_DIM + j] + az)));
      float r = 1.f / (1.f + __expf(-(Xr[t * H_DIM + j] + ar)));
      zbuf[j]  = z;
      rhbuf[j] = r * h[j];
    }
    grid_barrier(bar, bar + 1, REC_BLOCKS);
    // phase B: candidate row on (r .* h)
    float ah = 0.f;
    for (int k = lane; k < H_DIM; k += 32) ah += wh[k] * rhbuf[k];
    ah = wave_sum(ah);
    if (lane == 0) {
      float hp = tanhf(Xh[t * H_DIM + j] + ah);
      float z  = zbuf[j];
      Hbuf[(long long)(t + 1) * H_DIM + j] = (1.f - z) * h[j] + z * hp;
    }
    grid_barrier(bar, bar + 1, REC_BLOCKS);
  }
}

// ---------------------------------------------------- vocab projection (WMMA f32)
// One block per 16-wide vocab strip. 8 waves; wave w computes m-tiles {w, w+8}.
// B strip (16 x 1024 of out_w) is staged into LDS in two K-halves with
// CDNA5 async-to-LDS copies; each LDS B fetch feeds two WMMAs.
__global__ __launch_bounds__(256) void logits_wmma_kernel(
    const float* __restrict__ Hbuf, const float* __restrict__ out_w,
    const float* __restrict__ out_b, float* __restrict__ out) {
  __shared__ float Bs[16 * BSTRIDE];                 // 33 KB

  int colbase = blockIdx.x * 16;
  int tid  = threadIdx.x;
  int w    = tid >> 5;                               // wave 0..7
  int lane = tid & 31;
  int mn   = lane & 15;
  int khalf = (lane >> 4) << 1;
  int mhi  = (lane >> 4) * 8;
  int col  = colbase + mn;

  int mt0 = w, mt1 = w + 8;
  const float* A0base = Hbuf + (long long)(mt0 * 16 + mn + 1) * H_DIM;
  const float* A1base = Hbuf + (long long)(mt1 * 16 + mn + 1) * H_DIM;

  float cinit = out_b[col];
  v8f c0, c1;
#pragma unroll
  for (int i = 0; i < 8; ++i) { c0[i] = cinit; c1[i] = cinit; }

  // staging role: 16 threads per B row
  int r  = tid >> 4;                                 // row 0..15
  int ct = tid & 15;
  uint32_t ldsrow = (uint32_t)(size_t)(&Bs[r * BSTRIDE]);   // LDS byte offset (low 32 bits)
  const float* bptr = &Bs[mn * BSTRIDE + khalf];

  for (int half = 0; half < 2; ++half) {
    // ---- async-stage 16 x 512 floats of out_w into LDS (32 KB) ----
    const float* grow = out_w + (long long)(colbase + r) * H_DIM + half * BK;
#pragma unroll
    for (int it = 0; it < BK / (16 * 4); ++it) {     // 8 x b128 per thread
      int cc = (ct + it * 16) * 4;                   // float index of 16B chunk
      uint64_t ga = (uint64_t)(size_t)(grow + cc);
      uint32_t la = ldsrow + (uint32_t)(cc * 4);
      asm volatile("global_load_async_to_lds_b128 %0, %1, off"
                   :: "v"(la), "v"(ga) : "memory");
    }
    asm volatile("s_wait_asynccnt 0x0" ::: "memory");
    __syncthreads();

    // ---- WMMA over this K-half: B from LDS, two m-tiles share each B fetch ----
    const float* A0 = A0base + half * BK;
    const float* A1 = A1base + half * BK;
#pragma unroll 4
    for (int k = 0; k < BK; k += 4) {
      v2f b  = *(const v2f*)(bptr + k);              // ds_load_b64, conflict-free
      v2f a0 = *(const v2f*)(A0 + k + khalf);
      v2f a1 = *(const v2f*)(A1 + k + khalf);
      c0 = __builtin_amdgcn_wmma_f32_16x16x4_f32(false, a0, false, b, (short)0, c0, false, false);
      c1 = __builtin_amdgcn_wmma_f32_16x16x4_f32(false, a1, false, b, (short)0, c1, false, false);
    }
    __syncthreads();                                 // before re-staging LDS
  }

#pragma unroll
  for (int i = 0; i < 8; ++i) {
    out[(long long)(mt0 * 16 + i + mhi) * V_DIM + col] = c0[i];
    out[(long long)(mt1 * 16 + i + mhi) * V_DIM + col] = c1[i];
  }
}

// ---------------------------------------------------------------- argmax (first max)
__global__ __launch_bounds__(256) void argmax_kernel(const float* __restrict__ logits,
                                                     int* __restrict__ pred) {
  int t = blockIdx.x;
  const float* row = logits + (long long)t * V_DIM;
  int tid = threadIdx.x;
  float best = -INFINITY; int bidx = V_DIM;
  for (int v = tid; v < V_DIM; v += 256) {
    float x = row[v];
    if (x > best) { best = x; bidx = v; }        // strided ascending -> keeps lowest idx
  }
  __shared__ float sv[256];
  __shared__ int   si[256];
  sv[tid] = best; si[tid] = bidx;
  __syncthreads();
  for (int s = 128; s > 0; s >>= 1) {
    if (tid < s) {
      if (sv[tid + s] > sv[tid] || (sv[tid + s] == sv[tid] && si[tid + s] < si[tid])) {
        sv[tid] = sv[tid + s]; si[tid] = si[tid + s];
      }
    }
    __syncthreads();
  }
  if (tid == 0) pred[t] = si[0];
}

// ---------------------------------------------------------------- launch
extern "C" void kernel_launch(void* const* d_in, const int* in_sizes, int n_in,
                              void* d_out, int out_size, void* d_ws, size_t ws_size,
                              hipStream_t stream) {
  const float* encoding  = (const float*)d_in[0];
  const int*   tokens    = (const int*)d_in[1];
  const float* embedding = (const float*)d_in[2];
  const float* wzx_w = (const float*)d_in[3];
  const float* wzx_b = (const float*)d_in[4];
  const float* wrx_w = (const float*)d_in[5];
  const float* wrx_b = (const float*)d_in[6];
  const float* whx_w = (const float*)d_in[7];
  const float* whx_b = (const float*)d_in[8];
  const float* wzh_w = (const float*)d_in[9];
  const float* wzh_b = (const float*)d_in[10];
  const float* wrh_w = (const float*)d_in[11];
  const float* wrh_b = (const float*)d_in[12];
  const float* whh_w = (const float*)d_in[13];
  const float* whh_b = (const float*)d_in[14];
  const float* out_w = (const float*)d_in[15];
  const float* out_b = (const float*)d_in[16];

  float* out  = (float*)d_out;
  int*   pred = (int*)d_out + (long long)T_STEPS * V_DIM;   // int32 tail of d_out

  // workspace layout (floats)
  float* ws    = (float*)d_ws;
  float* Xz    = ws;                                   // T*H
  float* Xr    = Xz + T_STEPS * H_DIM;                 // T*H
  float* Xh    = Xr + T_STEPS * H_DIM;                 // T*H
  float* Hbuf  = Xh + T_STEPS * H_DIM;                 // (T+1)*H, row 0 = encoding
  float* zbuf  = Hbuf + (T_STEPS + 1) * H_DIM;         // H
  float* rhbuf = zbuf + H_DIM;                         // H
  int*   bar   = (int*)(rhbuf + H_DIM);                // 2 ints

  init_kernel<<<1, 256, 0, stream>>>(encoding, Hbuf, bar);

  // 3 gates * (16 m-tiles * 64 n-tiles) = 3072 waves = 384 blocks
  x_precompute_kernel<<<384, 256, 0, stream>>>(tokens, embedding,
      wzx_w, wzx_b, wrx_w, wrx_b, whx_w, whx_b,
      wzh_b, wrh_b, whh_b, Xz, Xr, Xh);

  gru_recurrent_kernel<<<REC_BLOCKS, 256, 0, stream>>>(Xz, Xr, Xh,
      wzh_w, wrh_w, whh_w, Hbuf, zbuf, rhbuf, bar);

  // one block per 16-wide vocab strip: 2000 blocks
  logits_wmma_kernel<<<V_DIM / 16, 256, 0, stream>>>(Hbuf, out_w, out_b, out);

  argmax_kernel<<<T_STEPS, 256, 0, stream>>>(out, pred);
}